// SORDefense_68247030334077
// MI455X (gfx1250) — compile-verified
//
#include <hip/hip_runtime.h>
#include <math.h>

#define NB    4
#define NPTS  8192
#define ALPHA 1.1f

typedef float v2f __attribute__((ext_vector_type(2)));
typedef float v8f __attribute__((ext_vector_type(8)));

__device__ __forceinline__ float med3f(float a, float b, float c) {
#if __has_builtin(__builtin_amdgcn_fmed3f)
  return __builtin_amdgcn_fmed3f(a, b, c);   // v_med3_num_f32, no canonicalize ops
#else
  return fmaxf(fminf(a, b), fminf(fmaxf(a, b), c));
#endif
}

// Insert d into sorted triple s0 <= s1 <= s2 (keep 3 smallest): 3 med3 ops.
//   s0' = min(d,s0) = med3(d,s0,-BIG); s1' = med3(d,s0,s1); s2' = med3(d,s1,s2)
__device__ __forceinline__ void insert3(float d, float& s0, float& s1, float& s2) {
  float n0 = med3f(d, s0, -3.0e38f);
  float n1 = med3f(d, s0, s1);
  float n2 = med3f(d, s1, s2);
  s0 = n0; s1 = n1; s2 = n2;
}

// Kernel 1: per-point mean-2NN distance. The WMMA f32 16x16x4 directly emits
// the shifted distance t = ||j||^2 - 2 i.j by packing A=(-2x,-2y,-2z,1) and
// B=(x,y,z,||j||^2) into the K=4 slots. grid.x = NB*(NPTS/128); 8 waves/block.
extern "C" __global__ __launch_bounds__(256)
void sor_knn_value(const float* __restrict__ x, float* __restrict__ value)
{
  extern __shared__ float4 pts[];          // {x,y,z,||p||^2}: 8192*16B = 128 KB LDS
  const int rowBlocks = NPTS / 128;
  const int b      = blockIdx.x / rowBlocks;
  const int rowblk = blockIdx.x % rowBlocks;
  const float* px  = x + (size_t)b * NPTS * 3;

  // Stage whole batch into LDS with precomputed squared norm (coalesced reads).
  for (int i = threadIdx.x; i < NPTS; i += 256) {
    float x0 = px[3*i+0], y0 = px[3*i+1], z0 = px[3*i+2];
    float4 p;
    p.x = x0; p.y = y0; p.z = z0;
    p.w = fmaf(x0, x0, fmaf(y0, y0, z0*z0));
    pts[i] = p;
  }
  __syncthreads();

  const int  lane = threadIdx.x & 31;      // wave32
  const int  wave = threadIdx.x >> 5;
  const int  lmod = lane & 15;
  const bool hiH  = lane >= 16;
  const int  rowbase = rowblk * 128 + wave * 16;
  const int  rbase   = rowbase + (hiH ? 8 : 0);   // C/D layout: lanes>=16 hold M=r+8

  // A-matrix (16x4 f32, ISA layout): lanes 0-15 hold (K0,K1)=(-2x,-2y) of row
  // M=lane, lanes 16-31 hold (K2,K3)=(-2z, 1). The unit K3 column pulls the
  // column norm out of B, so D = -2 i.j + ||j||^2 with zero epilogue VALU.
  const float4 pr = pts[rowbase + lmod];
  v2f A;
  A.x = hiH ? (-2.0f * pr.z) : (-2.0f * pr.x);
  A.y = hiH ? 1.0f           : (-2.0f * pr.y);

  // Top-3 smallest of t = ||j||^2 - 2 i.j; the per-row constant ||i||^2 is a
  // monotonic shift, restored at the end. Self wins s0 (t_j - t_self >= 0).
  float s0[8], s1[8], s2[8];
#pragma unroll
  for (int r = 0; r < 8; ++r) { s0[r] = 3.0e38f; s1[r] = 3.0e38f; s2[r] = 3.0e38f; }

  // B register pair is exactly one half of the staged float4: (x,y) for lanes
  // 0-15, (z,||p||^2) for lanes 16-31 -> a single conflict-free ds_load_b64.
  const float2* pts2 = reinterpret_cast<const float2*>(pts);
  const int bsel = hiH ? 1 : 0;

#pragma unroll 4
  for (int jt = 0; jt < NPTS/16; ++jt) {
    const int j = jt*16 + lmod;
#if defined(__gfx1250__) && __has_builtin(__builtin_amdgcn_wmma_f32_16x16x4_f32)
    const float2 bj = pts2[2*j + bsel];    // ds_load_b64, all 64 banks hit once
    v2f Bm;
    Bm.x = bj.x;
    Bm.y = bj.y;
    v8f C = {0.f,0.f,0.f,0.f,0.f,0.f,0.f,0.f};
    C = __builtin_amdgcn_wmma_f32_16x16x4_f32(false, A, false, Bm,
                                              (short)0, C, false, false);
#else
    const float4 pj = pts[j];
    v8f C;
#pragma unroll
    for (int r = 0; r < 8; ++r) {
      float4 q = pts[rbase + r];
      float dot = fmaf(q.x, pj.x, fmaf(q.y, pj.y, q.z * pj.z));
      C[r] = fmaf(-2.0f, dot, pj.w);
    }
#endif
#pragma unroll
    for (int r = 0; r < 8; ++r)
      insert3(C[r], s0[r], s1[r], s2[r]);  // WMMA output feeds selection directly
  }

  // Merge the 16 column-lane partial top-3s per row: xor-shuffle tree (masks
  // 1,2,4,8 stay inside each 16-lane half).
#pragma unroll
  for (int m = 1; m <= 8; m <<= 1) {
#pragma unroll
    for (int r = 0; r < 8; ++r) {
      float t0 = __shfl_xor(s0[r], m, 32);
      float t1 = __shfl_xor(s1[r], m, 32);
      float t2 = __shfl_xor(s2[r], m, 32);
      insert3(t0, s0[r], s1[r], s2[r]);
      insert3(t1, s0[r], s1[r], s2[r]);
      insert3(t2, s0[r], s1[r], s2[r]);
    }
  }

  // Drop self (s0); value = mean of 2 NN distances, restoring the deferred
  // row norm: 0.5*((s1+rxx)+(s2+rxx)) = 0.5*(s1+s2) + rxx. Lanes 0/16 write.
  if (lmod == 0) {
    float* v = value + b * NPTS + rbase;
#pragma unroll
    for (int r = 0; r < 8; ++r) {
      float rxx = pts[rbase + r].w;        // LDS still holds the staged points
      v[r] = fmaf(0.5f, s1[r] + s2[r], rxx);
    }
  }
}

// Kernel 2: per-batch mean/std(ddof=1) -> threshold. One block per batch.
extern "C" __global__ __launch_bounds__(256)
void sor_thresh(const float* __restrict__ value, float* __restrict__ thr)
{
  __shared__ float ssum[256];
  __shared__ float ssq[256];
  const int b = blockIdx.x;
  const float* v = value + b * NPTS;
  float s = 0.0f, q = 0.0f;
  for (int i = threadIdx.x; i < NPTS; i += 256) {
    float t = v[i];
    s += t;
    q = fmaf(t, t, q);
  }
  ssum[threadIdx.x] = s; ssq[threadIdx.x] = q;
  __syncthreads();
  for (int off = 128; off > 0; off >>= 1) {
    if (threadIdx.x < off) {
      ssum[threadIdx.x] += ssum[threadIdx.x + off];
      ssq[threadIdx.x]  += ssq[threadIdx.x + off];
    }
    __syncthreads();
  }
  if (threadIdx.x == 0) {
    float mean = ssum[0] * (1.0f / NPTS);
    float var  = (ssq[0] - (float)NPTS * mean * mean) * (1.0f / (NPTS - 1));
    var = fmaxf(var, 0.0f);
    thr[b] = fmaf(ALPHA, sqrtf(var), mean);
  }
}

// Kernel 3: apply mask; out = [masked_pc (B*K*3), mask-as-float (B*K)].
extern "C" __global__ __launch_bounds__(256)
void sor_mask(const float* __restrict__ x, const float* __restrict__ value,
              const float* __restrict__ thr, float* __restrict__ out)
{
  const int idx = blockIdx.x * 256 + threadIdx.x;   // global point index
  if (idx >= NB * NPTS) return;
  const int b = idx / NPTS;
  const float m = (value[idx] <= thr[b]) ? 1.0f : 0.0f;
  const float* xp = x + (size_t)idx * 3;
  float* op = out + (size_t)idx * 3;
  op[0] = xp[0] * m;
  op[1] = xp[1] * m;
  op[2] = xp[2] * m;
  out[(size_t)NB * NPTS * 3 + idx] = m;
}

extern "C" void kernel_launch(void* const* d_in, const int* in_sizes, int n_in,
                              void* d_out, int out_size, void* d_ws, size_t ws_size,
                              hipStream_t stream) {
  (void)in_sizes; (void)n_in; (void)out_size; (void)ws_size;
  const float* x = (const float*)d_in[0];        // [4, 8192, 3] fp32
  float* out   = (float*)d_out;                  // 98304 + 32768 floats
  float* value = (float*)d_ws;                   // NB*NPTS floats scratch
  float* thr   = value + NB * NPTS;              // NB floats scratch

  dim3 blk(256, 1, 1);
  dim3 g1(NB * (NPTS / 128), 1, 1);              // 256 workgroups, 8 waves each
  size_t shmem = NPTS * sizeof(float4);          // 128 KB LDS per workgroup
  sor_knn_value<<<g1, blk, shmem, stream>>>(x, value);
  sor_thresh<<<dim3(NB, 1, 1), blk, 0, stream>>>(value, thr);
  const int n = NB * NPTS;
  sor_mask<<<dim3((n + 255) / 256, 1, 1), blk, 0, stream>>>(x, value, thr, out);
}